// MultiVectorQuantizer_68994354643080
// MI455X (gfx1250) — compile-verified
//
#include <hip/hip_runtime.h>
#include <math.h>

typedef float v2f __attribute__((ext_vector_type(2)));
typedef float v8f __attribute__((ext_vector_type(8)));

#define CC    64          // channels (GEMM contraction dim)
#define HW    1024        // 32*32
#define NN    131072      // 128 * 1024 spatial positions
#define KK    256         // codebook size
#define ZQ_ELEMS 8388608  // 128*64*32*32
// d_out layout: [0,8388608) z_q | 8388608..8388610 scalars | 8388611.. indices(float)
#define OUT_SCALARS 8388608
#define OUT_IDX     8388611
// ws layout (floats): [0]=S_z acc, [1]=loss acc, [64..319]=e_sq[256], [320..)=idx as int
#define WS_ESQ 64
#define WS_IDX 320

__global__ __launch_bounds__(256)
void vq_init(const float* __restrict__ E, float* __restrict__ wsf) {
    int k = threadIdx.x;            // 256 threads, one per code
    if (k < 2) wsf[k] = 0.0f;
    float s = 0.0f;
    #pragma unroll 8
    for (int c = 0; c < CC; ++c) { float e = E[k * CC + c]; s += e * e; }
    wsf[WS_ESQ + k] = s;
}

__global__ __launch_bounds__(256)
void vq_main(const float* __restrict__ z1, const float* __restrict__ z2,
             const float* __restrict__ z3, const float* __restrict__ E,
             float* __restrict__ out, float* __restrict__ wsf,
             int* __restrict__ idxOut) {
    // 64KB of LDS: B fragments in WMMA layout.
    // frag(kstep,ktile) at float offset (kstep*16+ktile)*64 + lane*2 + v
    // element: K-comp = v + 2*(lane>=16), code col = ktile*16 + lane%16,
    //          c = kstep*4 + K-comp, value = E[code][c]
    __shared__ float sB[16384];

    const int tid = threadIdx.x;
    for (int g = tid; g < 16384; g += 256) {
        int frag   = g >> 6;
        int within = g & 63;
        int fl     = within >> 1;
        int v      = within & 1;
        int kstep  = frag >> 4;
        int ktile  = frag & 15;
        int c      = kstep * 4 + v + ((fl >= 16) ? 2 : 0);
        int code   = ktile * 16 + (fl & 15);
        sB[g] = E[code * CC + c];
    }
    __syncthreads();

    const int lane = tid & 31;
    const int wave = tid >> 5;
    const int half = lane >> 4;                       // 0: K=0..1, rows M=v ; 1: K=2..3, rows M=v+8
    const int rowbase = blockIdx.x * 128 + wave * 16; // wave strip: 16 rows
    const int row = rowbase + (lane & 15);            // n for this lane's A rows
    const int bimg = row >> 10;                       // row / HW  (strip never crosses b)
    const int hw   = row & 1023;
    const long base = (long)bimg * (CC * HW) + hw;    // element offset at c=0 in BCHW

    // ---- Build A fragments (zsum in WMMA A layout) + per-element square sums ----
    v2f afrag[16];
    float ssq = 0.0f;
    #pragma unroll
    for (int ks = 0; ks < 16; ++ks) {
        const int c0 = ks * 4 + 2 * half;
        #pragma unroll
        for (int v = 0; v < 2; ++v) {
            long off = base + (long)(c0 + v) * HW;    // lanes 0-15 contiguous in hw -> coalesced
            float a1 = z1[off], a2 = z2[off], a3 = z3[off];
            ssq += a1 * a1 + a2 * a2 + a3 * a3;
            afrag[ks][v] = a1 + a2 + a3;
        }
    }
    // wave-reduce S_z partial, one atomic per wave
    #pragma unroll
    for (int m = 16; m >= 1; m >>= 1) ssq += __shfl_xor(ssq, m, 32);
    if (lane == 0) atomicAdd(&wsf[0], ssq);

    // ---- GEMM + fused argmin over 256 codes ----
    float bestv[8];
    int   besti[8];
    #pragma unroll
    for (int v = 0; v < 8; ++v) { bestv[v] = __builtin_inff(); besti[v] = 0; }

    const float* __restrict__ esq = &wsf[WS_ESQ];
    for (int kt = 0; kt < 16; ++kt) {
        v8f acc = {0.f, 0.f, 0.f, 0.f, 0.f, 0.f, 0.f, 0.f};
        #pragma unroll
        for (int ks = 0; ks < 16; ++ks) {
            v2f bfrag = *(const v2f*)(&sB[(ks * 16 + kt) * 64 + lane * 2]);
            acc = __builtin_amdgcn_wmma_f32_16x16x4_f32(
                false, afrag[ks], false, bfrag, (short)0, acc, false, false);
        }
        const int col = kt * 16 + (lane & 15);
        const float eq = esq[col];
        #pragma unroll
        for (int v = 0; v < 8; ++v) {
            float s = eq - (2.0f / 3.0f) * acc[v];    // affine in true distance: same argmin & ties
            if (s < bestv[v]) { bestv[v] = s; besti[v] = col; }  // kt ascending -> first-min kept
        }
    }

    // ---- cross-lane argmin within each 16-lane half; tie -> lower index ----
    float lsum = 0.0f;
    #pragma unroll
    for (int v = 0; v < 8; ++v) {
        float bv = bestv[v]; int bi = besti[v];
        #pragma unroll
        for (int m = 8; m >= 1; m >>= 1) {
            float ov = __shfl_xor(bv, m, 32);
            int   oi = __shfl_xor(bi, m, 32);
            if (ov < bv || (ov == bv && oi < bi)) { bv = ov; bi = oi; }
        }
        if ((lane & 15) == 0) {
            int n = rowbase + half * 8 + v;           // half 0 -> rows v, half 1 -> rows v+8
            idxOut[n] = bi;
            out[OUT_IDX + n] = (float)bi;
            lsum += 3.0f * bv;                        // 3*score_best = 3*e_sq - 2*cross at argmin
        }
    }
    if ((lane & 15) == 0) atomicAdd(&wsf[1], lsum);
}

__global__ __launch_bounds__(256)
void vq_gather(const float* __restrict__ E, const int* __restrict__ idx,
               float* __restrict__ out) {
    int t  = blockIdx.x * 256 + threadIdx.x;          // over B*C*H*W in BCHW order
    int b  = t >> 16;                                 // t / (C*HW)
    int c  = (t >> 10) & 63;
    int hw = t & 1023;
    int n  = b * HW + hw;
    out[t] = E[idx[n] * CC + c];                      // coalesced write; E is L2-resident
}

__global__ void vq_final(const float* __restrict__ wsf, float* __restrict__ out) {
    // loss = (Sum_n 3*bestscore + S_z) / (3 * N * C)
    float L = (wsf[1] + wsf[0]) * (1.0f / 25165824.0f);
    out[OUT_SCALARS + 0] = L;          // codebook_loss
    out[OUT_SCALARS + 1] = L;          // commitment_loss (same value; stop_grad only affects grads)
    out[OUT_SCALARS + 2] = 1.25f * L;  // vq_loss = codebook + BETA*commitment, BETA=0.25
}

extern "C" void kernel_launch(void* const* d_in, const int* in_sizes, int n_in,
                              void* d_out, int out_size, void* d_ws, size_t ws_size,
                              hipStream_t stream) {
    const float* z1 = (const float*)d_in[0];
    const float* z2 = (const float*)d_in[1];
    const float* z3 = (const float*)d_in[2];
    const float* E  = (const float*)d_in[3];
    float* out = (float*)d_out;
    float* wsf = (float*)d_ws;
    int*   idx = (int*)(wsf + WS_IDX);
    (void)in_sizes; (void)n_in; (void)out_size; (void)ws_size;

    vq_init  <<<1, 256, 0, stream>>>(E, wsf);
    vq_main  <<<NN / 128, 256, 0, stream>>>(z1, z2, z3, E, out, wsf, idx);
    vq_gather<<<ZQ_ELEMS / 256, 256, 0, stream>>>(E, idx, out);
    vq_final <<<1, 1, 0, stream>>>(wsf, out);
}